// OneCardPokerPaynet_57647051047443
// MI455X (gfx1250) — compile-verified
//
#include <hip/hip_runtime.h>

typedef __attribute__((ext_vector_type(2))) float v2f;
typedef __attribute__((ext_vector_type(4))) float f4;
typedef __attribute__((ext_vector_type(8))) float v8f;

#define NC 32

__global__ __launch_bounds__(256) void OneCardPokerPaynet_kernel(
    const float* __restrict__ x,
    const float* __restrict__ Wfc,
    const float* __restrict__ Wdist,
    float* __restrict__ P_out,
    float* __restrict__ j_out,
    float* __restrict__ r_out)
{
    __shared__ float sR[NC];         // r[k]
    __shared__ float sU[NC];         // r[i]/(1-r[i])
    __shared__ float sQ[NC][NC];     // Q[i][k] = u_i * r_k  (from WMMA)
    __shared__ float sJ[2];          // {initial, traise}

    const int tid  = threadIdx.x;
    const int lane = tid & 31;
    const int wave = tid >> 5;
    const int b    = blockIdx.x;
    const float xb = x[b];

    // ---- Phase 1: softmax over 32 cards (wave 0 only) ----
    if (wave == 0) {
        const float logit = xb * Wdist[lane];
        float m = logit;
        #pragma unroll
        for (int off = 16; off > 0; off >>= 1)
            m = fmaxf(m, __shfl_xor(m, off));
        const float e = __expf(logit - m);
        float s = e;
        #pragma unroll
        for (int off = 16; off > 0; off >>= 1)
            s += __shfl_xor(s, off);
        const float r = e / s;
        sR[lane] = r;
        sU[lane] = r / (1.0f - r);
        r_out[(size_t)b * NC + lane] = r;
        if (lane == 0) {
            const float j0 = xb * Wfc[0];
            const float j1 = xb * Wfc[1];
            j_out[2 * (size_t)b + 0] = j0;
            j_out[2 * (size_t)b + 1] = j1;
            sJ[0] = j0;        // initial
            sJ[1] = j0 + j1;   // traise
        }
    }
    __syncthreads();

    // ---- Phase 2: Q = u ⊗ r via V_WMMA_F32_16X16X4_F32 (rank-1, K=0 row) ----
    // Four 16x16 tiles: wave w in [0,4) handles tile (I = w>>1, J = w&1).
    if (wave < 4) {
        const int I = wave >> 1;
        const int J = wave & 1;
        // 32-bit A 16x4 layout: lanes 0-15 hold M=lane; VGPR0 = K=0 (K=2 for
        // lanes 16-31), VGPR1 = K=1/3.  Only K=0 is nonzero -> rank-1 product,
        // so B's K=1..3 slots are don't-care (we zero them anyway).
        v2f A;
        A.x = (lane < 16) ? sU[I * 16 + lane] : 0.0f;
        A.y = 0.0f;
        v2f Bm;
        Bm.x = (lane < 16) ? sR[J * 16 + lane] : 0.0f;
        Bm.y = 0.0f;
        v8f C = {};
        v8f D = __builtin_amdgcn_wmma_f32_16x16x4_f32(
            false, A, false, Bm, (short)0, C, false, false);
        // D layout: lane l, VGPR v -> (M = v + 8*(l>=16), N = l&15)
        const int Mb = I * 16 + ((lane >> 4) << 3);
        const int N  = J * 16 + (lane & 15);
        #pragma unroll
        for (int v = 0; v < 8; ++v)
            sQ[Mb + v][N] = D[v];
    }
    __syncthreads();

    // ---- Phase 3: stream P out, one 512B row per wave-iteration ----
    const float ini = sJ[0];
    const float tra = sJ[1];
    float* __restrict__ Pb = P_out + (size_t)b * (128 * 128);

    #pragma unroll 4
    for (int t = 0; t < 16; ++t) {
        const int row = wave * 16 + t;
        const int i   = (row & 63) >> 1;   // card index for this row
        const int hr  = row >> 6;          // row half
        const int pr  = row & 1;           // row parity
        f4 out;
        #pragma unroll
        for (int e = 0; e < 4; ++e) {
            const int c  = 4 * lane + e;
            const int k  = (c & 63) >> 1;  // card index for this col
            const int hc = c >> 6;
            const int pc = c & 1;
            const float q   = sQ[i][k];
            const float sgn = (k > i) ? 1.0f : ((k < i) ? -1.0f : 0.0f);
            float val = 0.0f;
            if (hr == 0) {
                if (pr == 0) {
                    // P[0:64:2, 0:64:2] = cmp * Pi
                    if (hc == 0 && pc == 0) val = sgn * ini * q;
                } else {
                    // P[1:64:2, 64:128:2] = -Pi ; P[1:64:2, 65:128:2] = cmp * Pt
                    if (hc == 1) {
                        if (pc == 0) val = (k == i) ? 0.0f : -(ini * q);
                        else         val = sgn * tra * q;
                    }
                }
            } else {
                if (pr == 0) {
                    // P[64:128:2, 1:64:2] = Pi
                    if (hc == 0 && pc == 1) val = (k == i) ? 0.0f : (ini * q);
                } else {
                    // P[65:128:2, 1:64:2] = cmp * Pt
                    if (hc == 0 && pc == 1) val = sgn * tra * q;
                }
            }
            out[e] = val;
        }
        // Non-temporal: 536MB stream >> 192MB L2, no reuse.
        __builtin_nontemporal_store(out, (f4*)(Pb + (size_t)row * 128 + 4 * lane));
    }
}

extern "C" void kernel_launch(void* const* d_in, const int* in_sizes, int n_in,
                              void* d_out, int out_size, void* d_ws, size_t ws_size,
                              hipStream_t stream) {
    const float* x     = (const float*)d_in[0];
    const float* Wfc   = (const float*)d_in[1];
    const float* Wdist = (const float*)d_in[2];

    const int B = in_sizes[0];           // x is (B,1) -> B elements
    float* out  = (float*)d_out;
    float* P    = out;                                   // B * 128 * 128
    float* j    = out + (size_t)B * 128 * 128;           // B * 2
    float* r    = j + (size_t)B * 2;                     // B * 32

    OneCardPokerPaynet_kernel<<<B, 256, 0, stream>>>(x, Wfc, Wdist, P, j, r);
}